// CircuitGNN_50903952392201
// MI455X (gfx1250) — compile-verified
//
#include <hip/hip_runtime.h>

typedef __attribute__((ext_vector_type(2))) float v2f;
typedef __attribute__((ext_vector_type(8))) float v8f;

// single-instruction ReLU: v_med3_f32(x, 0, +inf) clamps to [0, inf)
__device__ __forceinline__ float relu1(float x) {
    return __builtin_amdgcn_fmed3f(x, 0.0f, __builtin_inff());
}

// ---------------------------------------------------------------------------
// Zero a float region
// ---------------------------------------------------------------------------
__global__ void zero_kernel(float* __restrict__ p, long n) {
    long t = (long)blockIdx.x * blockDim.x + threadIdx.x;
    if (t < n) p[t] = 0.0f;
}

// ---------------------------------------------------------------------------
// Degree: deg[dst[e]] += 1  (float atomics; dst = edge_index[1][e])
// ---------------------------------------------------------------------------
__global__ void deg_kernel(const long long* __restrict__ ei, float* __restrict__ deg,
                           long E) {
    long e = (long)blockIdx.x * blockDim.x + threadIdx.x;
    if (e < E) {
        long d = (long)ei[E + e];
        atomicAdd(&deg[d], 1.0f);
    }
}

// dinv[i] = rsqrt(deg[i] + 1)   (+1 for self loop; always > 0)
__global__ void dinv_kernel(const float* __restrict__ deg, float* __restrict__ dinv,
                            long N) {
    long i = (long)blockIdx.x * blockDim.x + threadIdx.x;
    if (i < N) dinv[i] = rsqrtf(deg[i] + 1.0f);
}

// ---------------------------------------------------------------------------
// Per-edge record: {src, dst, norm} packed in a float4 (one broadcast load
// per scatter thread instead of 2x int64 loads + 2 random dinv gathers).
// ---------------------------------------------------------------------------
__global__ void edge_prep_kernel(const long long* __restrict__ ei,
                                 const float* __restrict__ dinv,
                                 float4* __restrict__ rec, long E) {
    long e = (long)blockIdx.x * blockDim.x + threadIdx.x;
    if (e >= E) return;
    long s = (long)ei[e];
    long d = (long)ei[E + e];
    float4 r;
    r.x = __int_as_float((int)s);
    r.y = __int_as_float((int)d);
    r.z = dinv[s] * dinv[d];
    r.w = 0.0f;
    rec[e] = r;
}

// ---------------------------------------------------------------------------
// Layer-1 GEMM (K=3) with fused self-loop+bias epilogue:
//   h_raw[n][j] = sum_{k<3} x[n][k]*W1[k][j]
//   h_agg[n][j] = dinv[n]^2 * h_raw[n][j] + b[j]
// ---------------------------------------------------------------------------
__global__ void gemm_in3_kernel(const float* __restrict__ x, const float* __restrict__ W1,
                                const float* __restrict__ b, const float* __restrict__ dinv,
                                float* __restrict__ h_raw, float* __restrict__ h_agg,
                                long N) {
    long t = (long)blockIdx.x * blockDim.x + threadIdx.x;   // t < N*64
    if (t >= N * 64) return;
    long n = t >> 6;
    int  j = (int)(t & 63);
    const float* xr = x + n * 3;
    float v = xr[0] * W1[j] + xr[1] * W1[64 + j] + xr[2] * W1[128 + j];
    h_raw[t] = v;
    float di = dinv[n];
    h_agg[t] = di * di * v + b[j];
}

// ---------------------------------------------------------------------------
// 64x64 GEMM via V_WMMA_F32_16X16X4_F32 (fp32 exact), ReLU on input fused
// (v_med3 single-op), self-loop+bias epilogue fused (writes h_raw AND h_agg;
// h_agg may alias the input buffer: each wave fully consumes its own rows
// before writing them, and GEMM rows are independent).
//
// W staged in LDS TRANSPOSED with stride 66 so each per-lane B fragment
// (W[k0][n], W[k0+1][n]) is one contiguous 8B-aligned pair -> ds_load_b64
// (compiler fuses pairs into ds_load_2addr_b64), no register repacking;
// stride 66 keeps banks spread ((2n+k) mod 64).
// ---------------------------------------------------------------------------
__global__ __launch_bounds__(256) void gemm64_wmma_kernel(
    const float* __restrict__ in, const float* __restrict__ W,
    const float* __restrict__ bias, const float* __restrict__ dinv,
    float* __restrict__ h_raw, float* __restrict__ h_agg, long N) {

    __shared__ float sWT[64 * 66];   // sWT[n*66 + k] = W[k*64 + n]
    __shared__ float sB[64];
    for (int i = threadIdx.x; i < 64 * 64; i += 256) {
        int k = i >> 6, n = i & 63;
        sWT[n * 66 + k] = W[i];
    }
    if (threadIdx.x < 64) sB[threadIdx.x] = bias[threadIdx.x];
    __syncthreads();

    const int wave = threadIdx.x >> 5;      // 0..7
    const int lane = threadIdx.x & 31;
    const int m    = lane & 15;
    const int hi   = lane >> 4;             // 0 or 1
    const long nodeBase = ((long)blockIdx.x * 8 + wave) * 16;
    if (nodeBase >= N) return;              // wave-uniform: EXEC stays all-1s

    if (nodeBase + 16 > N) {
        // scalar tail (at most one wave in the whole grid), epilogue fused
        long n = nodeBase + m;
        if (hi == 0 && n < N) {
            float di = dinv[n];
            float di2 = di * di;
            for (int j = 0; j < 64; ++j) {
                float acc = 0.0f;
                for (int k = 0; k < 64; ++k) {
                    float xv = relu1(in[n * 64 + k]);
                    acc += xv * sWT[j * 66 + k];   // sWT[n][k]=W[k][n] -> W[k][j]
                }
                h_raw[n * 64 + j] = acc;
                h_agg[n * 64 + j] = di2 * acc + sB[j];
            }
        }
        return;
    }

    const float* rowp = in + (nodeBase + m) * 64;   // this lane's A row
    const float* bp0 = sWT + (m     ) * 66;         // B columns for the 4 n-tiles
    const float* bp1 = sWT + (m + 16) * 66;
    const float* bp2 = sWT + (m + 32) * 66;
    const float* bp3 = sWT + (m + 48) * 66;

    v8f c0 = {}, c1 = {}, c2 = {}, c3 = {};

    #pragma unroll
    for (int kk = 0; kk < 16; ++kk) {
        const int k0 = kk * 4 + 2 * hi;             // contiguous K pair per lane
        v2f a = *(const v2f*)(rowp + k0);
        a.x = relu1(a.x);
        a.y = relu1(a.y);

        const v2f b0 = *(const v2f*)(bp0 + k0);
        const v2f b1 = *(const v2f*)(bp1 + k0);
        const v2f b2 = *(const v2f*)(bp2 + k0);
        const v2f b3 = *(const v2f*)(bp3 + k0);

        c0 = __builtin_amdgcn_wmma_f32_16x16x4_f32(false, a, false, b0, (short)0, c0, false, false);
        c1 = __builtin_amdgcn_wmma_f32_16x16x4_f32(false, a, false, b1, (short)0, c1, false, false);
        c2 = __builtin_amdgcn_wmma_f32_16x16x4_f32(false, a, false, b2, (short)0, c2, false, false);
        c3 = __builtin_amdgcn_wmma_f32_16x16x4_f32(false, a, false, b3, (short)0, c3, false, false);
    }

    // C/D layout: lane l, VGPR v -> row M = v + 8*hi, col = (l&15) + 16*nt
    float* outR = h_raw + nodeBase * 64;
    float* outA = h_agg + nodeBase * 64;
    #pragma unroll
    for (int v = 0; v < 8; ++v) {
        const int rowM = v + 8 * hi;
        const float di = dinv[nodeBase + rowM];
        const float di2 = di * di;
        const long r = (long)rowM * 64;
        const float t0 = c0[v], t1 = c1[v], t2 = c2[v], t3 = c3[v];
        outR[r + m]      = t0;
        outR[r + m + 16] = t1;
        outR[r + m + 32] = t2;
        outR[r + m + 48] = t3;
        outA[r + m]      = di2 * t0 + sB[m];
        outA[r + m + 16] = di2 * t1 + sB[m + 16];
        outA[r + m + 32] = di2 * t2 + sB[m + 32];
        outA[r + m + 48] = di2 * t3 + sB[m + 48];
    }
}

// ---------------------------------------------------------------------------
// Edge scatter: out[dst] += norm * h[src]
// 16 threads per edge; one broadcast float4 record load + float4 h gather
// (coalesced 256B per edge) + 4 global f32 atomics.
// ---------------------------------------------------------------------------
__global__ void edge_scatter_kernel(const float4* __restrict__ rec,
                                    const float* __restrict__ h,
                                    float* __restrict__ out, long E) {
    long t = (long)blockIdx.x * blockDim.x + threadIdx.x;
    long e = t >> 4;
    if (e >= E) return;
    int q = (int)(t & 15) * 4;
    const float4 r = rec[e];
    long s = (long)__float_as_int(r.x);
    long d = (long)__float_as_int(r.y);
    float norm = r.z;
    const float4 v = *(const float4*)(h + s * 64 + q);
    float* op = out + d * 64 + q;
    atomicAdd(op + 0, norm * v.x);
    atomicAdd(op + 1, norm * v.y);
    atomicAdd(op + 2, norm * v.z);
    atomicAdd(op + 3, norm * v.w);
}

// ---------------------------------------------------------------------------
// Pool: sums[j] += sum_n relu(h[n][j]); block-local LDS reduction first.
// ---------------------------------------------------------------------------
__global__ __launch_bounds__(256) void pool_kernel(const float* __restrict__ h,
                                                   float* __restrict__ sums, long N) {
    __shared__ float acc[64];
    if (threadIdx.x < 64) acc[threadIdx.x] = 0.0f;
    __syncthreads();
    const int j   = threadIdx.x & 63;
    const int sub = threadIdx.x >> 6;           // 0..3
    long base = (long)blockIdx.x * 512;
    long end  = base + 512; if (end > N) end = N;
    float local = 0.0f;
    for (long n = base + sub; n < end; n += 4)
        local += relu1(h[n * 64 + j]);
    atomicAdd(&acc[j], local);
    __syncthreads();
    if (threadIdx.x < 64) atomicAdd(&sums[threadIdx.x], acc[threadIdx.x]);
}

// ---------------------------------------------------------------------------
// Final: out[j] = tanh( (sums/N) @ Wfc + bfc ),  j < 24
// ---------------------------------------------------------------------------
__global__ void final_kernel(const float* __restrict__ sums, const float* __restrict__ Wfc,
                             const float* __restrict__ bfc, float* __restrict__ out,
                             float invN) {
    int j = threadIdx.x;
    if (j < 24) {
        float acc = bfc[j];
        for (int k = 0; k < 64; ++k)
            acc += (sums[k] * invN) * Wfc[k * 24 + j];
        out[j] = tanhf(acc);
    }
}

// ---------------------------------------------------------------------------
extern "C" void kernel_launch(void* const* d_in, const int* in_sizes, int n_in,
                              void* d_out, int out_size, void* d_ws, size_t ws_size,
                              hipStream_t stream) {
    const float*     x   = (const float*)d_in[0];
    const long long* ei  = (const long long*)d_in[1];   // int64 [2, E]
    // d_in[2] = batch (all zeros) unused
    const float* W1  = (const float*)d_in[3];
    const float* b1  = (const float*)d_in[4];
    const float* W2  = (const float*)d_in[5];
    const float* b2  = (const float*)d_in[6];
    const float* W3  = (const float*)d_in[7];
    const float* b3  = (const float*)d_in[8];
    const float* Wfc = (const float*)d_in[9];
    const float* bfc = (const float*)d_in[10];
    float* out = (float*)d_out;

    const long N = in_sizes[0] / 3;
    const long E = in_sizes[1] / 2;

    // workspace (floats): [hA N*64][hB N*64][deg N][sums 64][dinv N][rec E*4]
    float*  ws   = (float*)d_ws;
    float*  hA   = ws;
    float*  hB   = hA + N * 64;
    float*  deg  = hB + N * 64;
    float*  sums = deg + N;
    float*  dinv = sums + 64;
    float4* rec  = (float4*)(dinv + N);    // 16B-aligned only for even N (offset (130N+64)*4 bytes)

    const int B = 256;
    const long NH = N * 64;
    dim3 gNH((unsigned)((NH + B - 1) / B));
    dim3 gN((unsigned)((N + B - 1) / B));
    dim3 gE((unsigned)((E + B - 1) / B));
    dim3 gE16((unsigned)((E * 16 + B - 1) / B));
    dim3 gW((unsigned)((N + 127) / 128));       // wmma: 8 waves/block * 16 nodes
    dim3 gP((unsigned)((N + 511) / 512));

    // degrees + dinv + edge records  (zero deg+sums region: N+64 floats)
    zero_kernel<<<dim3((unsigned)((N + 64 + B - 1) / B)), B, 0, stream>>>(deg, N + 64);
    deg_kernel<<<gE, B, 0, stream>>>(ei, deg, E);
    dinv_kernel<<<gN, B, 0, stream>>>(deg, dinv, N);
    edge_prep_kernel<<<gE, B, 0, stream>>>(ei, dinv, rec, E);

    // layer 1 (GEMM + self-loop/bias fused)
    gemm_in3_kernel<<<gNH, B, 0, stream>>>(x, W1, b1, dinv, hA, hB, N);
    edge_scatter_kernel<<<gE16, B, 0, stream>>>(rec, hA, hB, E);

    // layer 2: reads hB (ReLU fused), raw->hA, agg->hB in place
    gemm64_wmma_kernel<<<gW, B, 0, stream>>>(hB, W2, b2, dinv, hA, hB, N);
    edge_scatter_kernel<<<gE16, B, 0, stream>>>(rec, hA, hB, E);

    // layer 3
    gemm64_wmma_kernel<<<gW, B, 0, stream>>>(hB, W3, b3, dinv, hA, hB, N);
    edge_scatter_kernel<<<gE16, B, 0, stream>>>(rec, hA, hB, E);

    // pool (ReLU fused) + FC + tanh
    pool_kernel<<<gP, B, 0, stream>>>(hB, sums, N);
    final_kernel<<<dim3(1), 32, 0, stream>>>(sums, Wfc, bfc, out, 1.0f / (float)N);
}